// DSNTDoubleLoss_55035710931289
// MI455X (gfx1250) — compile-verified
//
#include <hip/hip_runtime.h>

typedef __attribute__((ext_vector_type(2))) float v2f;
typedef __attribute__((ext_vector_type(8))) float v8f;

#define HDIM 256
#define WDIM 256
#define HW   65536   // 256*256
#define NPAIR 256    // B*C = 32*8
#define NTHR  256    // 8 waves of 32

// One block per (b,c) pair.
//  Pass 1: block max of input (HBM-bound, float4 coalesced)
//  Pass 2: stable softmax sums S, Sx, Sy via WMMA ones-matrix accumulation
//          (input re-read hits L2: 192MB L2 >> 134MB working set)
//  Pass 3: argmax of target (value,index), first-occurrence tie-break
__global__ __launch_bounds__(NTHR) void dsnt_pair_kernel(
    const float* __restrict__ input,
    const float* __restrict__ target,
    float* __restrict__ ed_out) {
  const int pair = blockIdx.x;
  const int t    = threadIdx.x;
  const int lane = t & 31;
  const int wv   = t >> 5;

  const float* __restrict__ in = input  + (size_t)pair * HW;
  const float* __restrict__ tg = target + (size_t)pair * HW;

  __shared__ float sval[NTHR];
  __shared__ int   sidx[NTHR];
  __shared__ float spart[8][3];

  // ---------------- Pass 1: block max of input ----------------
  float lmax = -__builtin_inff();
  const float4* __restrict__ in4 = (const float4*)in;
  #pragma unroll 4
  for (int k = 0; k < 64; ++k) {
    float4 v = in4[t + k * NTHR];
    lmax = fmaxf(lmax, fmaxf(fmaxf(v.x, v.y), fmaxf(v.z, v.w)));
  }
  sval[t] = lmax;
  __syncthreads();
  for (int s = NTHR / 2; s > 0; s >>= 1) {
    if (t < s) sval[t] = fmaxf(sval[t], sval[t + s]);
    __syncthreads();
  }
  const float m = sval[0];
  __syncthreads();

  // ---------------- Pass 2: WMMA-accumulated softmax sums ----------------
  // Wave wv owns rows [wv*32, wv*32+32). Tiles are 4 rows x 16 cols.
  // Lane L -> column n = L&15 of the tile; lane half (L>>4) picks rows
  // {h0+2*half, h0+2*half+1}. Each element's coordinate weight is applied
  // before it enters the B operand, so the three accumulators hold the
  // (layout-order-agnostic) running column sums of e, e*xg, e*yg.
  v2f ones; ones[0] = 1.0f; ones[1] = 1.0f;
  v8f c_e = {}; v8f c_x = {}; v8f c_y = {};

  const int   n     = lane & 15;
  const int   half  = lane >> 4;
  const int   hbase = wv * 32;
  const float inv   = 1.0f / 256.0f;

  for (int rc = 0; rc < 8; ++rc) {
    const int r0 = hbase + rc * 4 + 2 * half;
    const int r1 = r0 + 1;
    const float yg0 = (float)(r0 + 1) * inv;
    const float yg1 = (float)(r1 + 1) * inv;
    const float* __restrict__ row0 = in + r0 * WDIM;
    const float* __restrict__ row1 = in + r1 * WDIM;
    #pragma unroll 4
    for (int wc = 0; wc < 16; ++wc) {
      const int   w  = wc * 16 + n;
      const float xg = (float)(w + 1) * inv;
      const float e0 = __expf(row0[w] - m);
      const float e1 = __expf(row1[w] - m);
      v2f be; be[0] = e0;       be[1] = e1;
      v2f bx; bx[0] = e0 * xg;  bx[1] = e1 * xg;
      v2f by; by[0] = e0 * yg0; by[1] = e1 * yg1;
      // D = ones(16x4) x B(4x16) + C  ->  rows of D = running column sums
      c_e = __builtin_amdgcn_wmma_f32_16x16x4_f32(false, ones, false, be,
                                                  (short)0, c_e, false, false);
      c_x = __builtin_amdgcn_wmma_f32_16x16x4_f32(false, ones, false, bx,
                                                  (short)0, c_x, false, false);
      c_y = __builtin_amdgcn_wmma_f32_16x16x4_f32(false, ones, false, by,
                                                  (short)0, c_y, false, false);
    }
  }

  // c[0] of lane L = colsum[n=L&15] (rows replicated). Sum the 16-lane group.
  float ps = c_e[0], px = c_x[0], py = c_y[0];
  #pragma unroll
  for (int off = 8; off >= 1; off >>= 1) {
    ps += __shfl_xor(ps, off, 32);
    px += __shfl_xor(px, off, 32);
    py += __shfl_xor(py, off, 32);
  }
  if (lane == 0) { spart[wv][0] = ps; spart[wv][1] = px; spart[wv][2] = py; }

  // ---------------- Pass 3: argmax of target ----------------
  float bv = -__builtin_inff();
  int   bi = 0x7fffffff;
  const float4* __restrict__ tg4 = (const float4*)tg;
  #pragma unroll 4
  for (int k = 0; k < 64; ++k) {
    const int i4 = t + k * NTHR;
    float4 v = tg4[i4];
    const int base = i4 * 4;
    if (v.x > bv || (v.x == bv && base + 0 < bi)) { bv = v.x; bi = base + 0; }
    if (v.y > bv || (v.y == bv && base + 1 < bi)) { bv = v.y; bi = base + 1; }
    if (v.z > bv || (v.z == bv && base + 2 < bi)) { bv = v.z; bi = base + 2; }
    if (v.w > bv || (v.w == bv && base + 3 < bi)) { bv = v.w; bi = base + 3; }
  }
  sval[t] = bv;
  sidx[t] = bi;
  __syncthreads();
  for (int s = NTHR / 2; s > 0; s >>= 1) {
    if (t < s) {
      const float ov = sval[t + s];
      const int   oi = sidx[t + s];
      if (ov > sval[t] || (ov == sval[t] && oi < sidx[t])) {
        sval[t] = ov; sidx[t] = oi;
      }
    }
    __syncthreads();
  }

  // ---------------- Finalize this pair ----------------
  if (t == 0) {
    float S = 0.f, SX = 0.f, SY = 0.f;
    #pragma unroll
    for (int i = 0; i < 8; ++i) {
      S += spart[i][0]; SX += spart[i][1]; SY += spart[i][2];
    }
    const float predx = SX / S;
    const float predy = SY / S;
    const int idx = sidx[0];
    const float tx = (float)((idx & (WDIM - 1)) + 1) * inv;
    const float ty = (float)((idx >> 8) + 1) * inv;
    const float dx = tx - predx;
    const float dy = ty - predy;
    ed_out[pair] = __builtin_sqrtf(dx * dx + dy * dy);
  }
}

// Deterministic final reduction over the 256 per-pair distances; /B (B=32).
__global__ __launch_bounds__(NPAIR) void dsnt_finalize_kernel(
    const float* __restrict__ ed, float* __restrict__ out) {
  __shared__ float s[NPAIR];
  const int t = threadIdx.x;
  s[t] = ed[t];
  __syncthreads();
  for (int k = NPAIR / 2; k > 0; k >>= 1) {
    if (t < k) s[t] += s[t + k];
    __syncthreads();
  }
  if (t == 0) out[0] = s[0] * (1.0f / 32.0f);
}

extern "C" void kernel_launch(void* const* d_in, const int* in_sizes, int n_in,
                              void* d_out, int out_size, void* d_ws, size_t ws_size,
                              hipStream_t stream) {
  const float* input  = (const float*)d_in[0];
  const float* target = (const float*)d_in[1];
  float* out = (float*)d_out;
  float* ed  = (float*)d_ws;  // 256 floats of scratch, fully rewritten each call

  dsnt_pair_kernel<<<NPAIR, NTHR, 0, stream>>>(input, target, ed);
  dsnt_finalize_kernel<<<1, NPAIR, 0, stream>>>(ed, out);
}